// MultiHead_gat_68977174774337
// MI455X (gfx1250) — compile-verified
//
#include <hip/hip_runtime.h>
#include <hip/hip_bf16.h>
#include <cfloat>

typedef float v2f __attribute__((ext_vector_type(2)));
typedef float v8f __attribute__((ext_vector_type(8)));

#define D 64
#define H 4
#define HD 256   // H*D
#define NEG_SLOPE 0.2f
#define LN_EPS 1e-5f

__device__ __forceinline__ float2 ld2(const float* p) {
    return *(const float2*)p;
}

__device__ __forceinline__ void atomicMaxFloat(float* addr, float v) {
    // valid ordering trick when buffer initialized to -FLT_MAX
    if (v >= 0.0f) atomicMax((int*)addr, __float_as_int(v));
    else           atomicMin((unsigned int*)addr, (unsigned int)__float_as_int(v));
}

// ---------------------------------------------------------------------------
// K1: wv = h @ W_node + b_node   (N x 64) @ (64 x 256) -> (N x 256)
// One wave per 16x16 output tile. grid = (N/16, 16), block = 32.
// A frag (16x4 f32): lanes 0-15 M=0..15, lanes 16-31 M=0..15; VGPR pair holds
// K = 2*(lane>>4) + {0,1}.  B frag mirrors with N=lane&15.
// C frag: VGPR r -> (M = r + 8*(lane>>4), N = lane&15).
// ---------------------------------------------------------------------------
__global__ void wv_gemm(const float* __restrict__ h,
                        const float* __restrict__ Wn,
                        const float* __restrict__ bn,
                        float* __restrict__ wv, int N) {
    const int lane   = threadIdx.x;          // 0..31
    const int m      = blockIdx.x * 16 + (lane & 15);
    const int kHalf  = (lane >> 4) * 2;      // 0 or 2
    const int n      = blockIdx.y * 16 + (lane & 15);

    v8f acc = {};
    for (int k0 = 0; k0 < D; k0 += 4) {
        const int k = k0 + kHalf;
        v2f a, b;
        a.x = h[(size_t)m * D + k];
        a.y = h[(size_t)m * D + k + 1];
        b.x = Wn[(size_t)k * HD + n];
        b.y = Wn[(size_t)(k + 1) * HD + n];
        acc = __builtin_amdgcn_wmma_f32_16x16x4_f32(
            false, a, false, b, (short)0, acc, false, false);
    }
    const int rb = 8 * (lane >> 4);
    const float bb = bn[n];
    const int row0 = blockIdx.x * 16;
#pragma unroll
    for (int r = 0; r < 8; ++r)
        wv[(size_t)(row0 + rb + r) * HD + n] = acc[r] + bb;
}

// ---------------------------------------------------------------------------
// K0: init smax=-inf, denom=0, agg=0
// ---------------------------------------------------------------------------
__global__ void init_ws(float* __restrict__ smax, float* __restrict__ denom,
                        float* __restrict__ agg, int N) {
    const int i = blockIdx.x * blockDim.x + threadIdx.x;
    if (i < N * H) { smax[i] = -FLT_MAX; denom[i] = 0.0f; }
    if (i < N * HD) agg[i] = 0.0f;
}

// ---------------------------------------------------------------------------
// K2: per-edge logits + leaky relu + segment max (atomic).
// One wave per edge; lane loads float2 of u,v per head; shfl-xor reduce.
// ---------------------------------------------------------------------------
__global__ void edge_logits(const float* __restrict__ wv,
                            const int* __restrict__ src,
                            const int* __restrict__ dst,
                            const float* __restrict__ att,
                            float* __restrict__ logit,
                            float* __restrict__ smax, int E) {
    const int lane = threadIdx.x & 31;
    const int e = blockIdx.x * (blockDim.x >> 5) + (threadIdx.x >> 5);
    if (e >= E) return;
    const int s = src[e], d = dst[e];
    const float* us = wv + (size_t)s * HD;
    const float* vs = wv + (size_t)d * HD;
#pragma unroll
    for (int hh = 0; hh < H; ++hh) {
        float2 u  = ld2(us  + hh * D + 2 * lane);
        float2 v  = ld2(vs  + hh * D + 2 * lane);
        float2 au = ld2(att + hh * (2 * D) + 2 * lane);
        float2 av = ld2(att + hh * (2 * D) + D + 2 * lane);
        float p = u.x * au.x + u.y * au.y + v.x * av.x + v.y * av.y;
#pragma unroll
        for (int off = 16; off; off >>= 1) p += __shfl_xor(p, off, 32);
        if (lane == 0) {
            float l = (p > 0.0f) ? p : NEG_SLOPE * p;
            logit[(size_t)e * H + hh] = l;
            atomicMaxFloat(&smax[(size_t)d * H + hh], l);
        }
    }
}

// ---------------------------------------------------------------------------
// K3: ex = exp(logit - smax[dst]); denom[dst] += ex; agg[dst] += ex * u
// Atomic f32 adds land in L2 (agg is 51MB < 192MB L2).
// ---------------------------------------------------------------------------
__global__ void edge_agg(const float* __restrict__ wv,
                         const int* __restrict__ src,
                         const int* __restrict__ dst,
                         const float* __restrict__ logit,
                         const float* __restrict__ smax,
                         float* __restrict__ denom,
                         float* __restrict__ agg, int E) {
    const int lane = threadIdx.x & 31;
    const int e = blockIdx.x * (blockDim.x >> 5) + (threadIdx.x >> 5);
    if (e >= E) return;
    const int s = src[e], d = dst[e];
    const float* us = wv + (size_t)s * HD;
    float* ag = agg + (size_t)d * HD;
#pragma unroll
    for (int hh = 0; hh < H; ++hh) {
        const float ex = __expf(logit[(size_t)e * H + hh] - smax[(size_t)d * H + hh]);
        if (lane == 0) atomicAdd(&denom[(size_t)d * H + hh], ex);
        float2 u = ld2(us + hh * D + 2 * lane);
        atomicAdd(&ag[hh * D + 2 * lane],     ex * u.x);
        atomicAdd(&ag[hh * D + 2 * lane + 1], ex * u.y);
    }
}

// ---------------------------------------------------------------------------
// K4: h_new = (agg/denom) @ W_scale + bias ; x = h_new + h ; LN ; ReLU
// Block = 128 threads (4 waves), each wave one 16x16 WMMA tile (K=256),
// tiles staged in LDS, then fused per-row LayerNorm+ReLU.
// ---------------------------------------------------------------------------
__global__ __launch_bounds__(128)
void out_gemm_ln(const float* __restrict__ agg,
                 const float* __restrict__ denom,
                 const float* __restrict__ Ws,
                 const float* __restrict__ bias,
                 const float* __restrict__ h,
                 const float* __restrict__ gamma,
                 const float* __restrict__ beta,
                 float* __restrict__ out, int N) {
    __shared__ float sx[16][68];     // 16 rows x 64 cols, padded stride
    __shared__ float smu[16], srs[16];

    const int t = threadIdx.x, wave = t >> 5, lane = t & 31;
    const int row0 = blockIdx.x * 16;
    const int m = row0 + (lane & 15);
    const int kHalf = (lane >> 4) * 2;
    const int n = wave * 16 + (lane & 15);

    float dinv[H];
#pragma unroll
    for (int hh = 0; hh < H; ++hh)
        dinv[hh] = 1.0f / fmaxf(denom[(size_t)m * H + hh], 1e-16f);

    v8f acc = {};
    for (int k0 = 0; k0 < HD; k0 += 4) {
        const int k = k0 + kHalf;
        const float di = dinv[k >> 6];   // (k,k+1) never straddle a head
        v2f a, b;
        a.x = agg[(size_t)m * HD + k] * di;
        a.y = agg[(size_t)m * HD + k + 1] * di;
        b.x = Ws[(size_t)k * D + n];
        b.y = Ws[(size_t)(k + 1) * D + n];
        acc = __builtin_amdgcn_wmma_f32_16x16x4_f32(
            false, a, false, b, (short)0, acc, false, false);
    }

    const int rb = 8 * (lane >> 4);
    const float bb = bias[n];
#pragma unroll
    for (int r = 0; r < 8; ++r) {
        const int mm = rb + r;
        sx[mm][n] = acc[r] + bb + h[(size_t)(row0 + mm) * D + n];
    }
    __syncthreads();

    // per-row mean/var: 8 threads per row, each sums 8 columns
    {
        const int row = t >> 3, c0 = (t & 7) * 8;
        float s = 0.0f, s2 = 0.0f;
#pragma unroll
        for (int c = 0; c < 8; ++c) {
            const float x = sx[row][c0 + c];
            s += x; s2 += x * x;
        }
#pragma unroll
        for (int off = 4; off; off >>= 1) {
            s  += __shfl_xor(s,  off, 32);
            s2 += __shfl_xor(s2, off, 32);
        }
        if ((t & 7) == 0) {
            const float mu = s * (1.0f / 64.0f);
            smu[row] = mu;
            srs[row] = rsqrtf(fmaxf(s2 * (1.0f / 64.0f) - mu * mu, 0.0f) + LN_EPS);
        }
    }
    __syncthreads();

    for (int idx = t; idx < 16 * D; idx += 128) {
        const int row = idx >> 6, col = idx & 63;
        const float y = (sx[row][col] - smu[row]) * srs[row] * gamma[col] + beta[col];
        out[(size_t)(row0 + row) * D + col] = fmaxf(y, 0.0f);
    }
}

// ---------------------------------------------------------------------------
extern "C" void kernel_launch(void* const* d_in, const int* in_sizes, int n_in,
                              void* d_out, int out_size, void* d_ws, size_t ws_size,
                              hipStream_t stream) {
    const float* h       = (const float*)d_in[0];
    const int*   src     = (const int*)  d_in[1];
    const int*   dst     = (const int*)  d_in[2];
    const float* W_node  = (const float*)d_in[3];
    const float* b_node  = (const float*)d_in[4];
    const float* att     = (const float*)d_in[5];
    const float* W_scale = (const float*)d_in[6];
    const float* bias    = (const float*)d_in[7];
    const float* gamma   = (const float*)d_in[8];
    const float* beta    = (const float*)d_in[9];
    float* out = (float*)d_out;

    const int N = in_sizes[0] / D;   // 50000
    const int E = in_sizes[1];       // 800000

    // workspace carve-out (floats): wv | logit | smax | denom | agg
    float* ws    = (float*)d_ws;
    float* wv    = ws;
    float* logit = wv    + (size_t)N * HD;
    float* smax  = logit + (size_t)E * H;
    float* denom = smax  + (size_t)N * H;
    float* agg   = denom + (size_t)N * H;

    // K1: node transform (WMMA f32)
    dim3 g1((N + 15) / 16, HD / 16);
    wv_gemm<<<g1, 32, 0, stream>>>(h, W_node, b_node, wv, N);

    // K0: init accumulators
    const int tot = N * HD;
    init_ws<<<(tot + 255) / 256, 256, 0, stream>>>(smax, denom, agg, N);

    // K2: logits + segment max (wave per edge, 8 edges per 256-thread block)
    edge_logits<<<(E + 7) / 8, 256, 0, stream>>>(wv, src, dst, att, logit, smax, E);

    // K3: softmax-weighted aggregation
    edge_agg<<<(E + 7) / 8, 256, 0, stream>>>(wv, src, dst, logit, smax, denom, agg, E);

    // K4: output transform + residual + LayerNorm + ReLU (WMMA f32, fused)
    out_gemm_ln<<<(N + 15) / 16, 128, 0, stream>>>(agg, denom, W_scale, bias, h,
                                                   gamma, beta, out, N);
}